// differentiableSuperpixelEmbedding_81432579932320
// MI455X (gfx1250) — compile-verified
//
#include <hip/hip_runtime.h>

#define MAX_SEG   196
#define MAX_PIX   400
#define NCH       3
#define EMBED     768
#define FANIN     (NCH * MAX_PIX)       // 1200
#define IMG_B     64
#define NPIX      (224 * 224)           // 50176
#define MROWS     (IMG_B * MAX_SEG)     // 12544

// GEMM tiling
#define BM        128
#define BN        128
#define KB        48                    // 1200 = 25 * 48
#define LDST      52                    // LDS row stride (floats): 16B-aligned float4, conflict-free frags

typedef __attribute__((ext_vector_type(2))) float v2f;
typedef __attribute__((ext_vector_type(8))) float v8f;

// ---------------------------------------------------------------------------
// CDNA5 async VMEM -> LDS copy (ASYNCcnt-tracked), per CDNA5 ISA
// ---------------------------------------------------------------------------
__device__ __forceinline__ void async_ld_b128(unsigned lds_byte_off, const void* gaddr) {
    asm volatile("global_load_async_to_lds_b128 %0, %1, off"
                 :: "v"(lds_byte_off), "v"(gaddr)
                 : "memory");
}
__device__ __forceinline__ void wait_async0() {
    asm volatile("s_wait_asynccnt 0" ::: "memory");
}

// ---------------------------------------------------------------------------
// Kernel 1: deterministic scan-order gather with truncation + fused tail-zero.
// One wave32 per (segment s, batch b). Ballot + prefix popcount reproduces the
// reference's stable-sort rank assignment exactly. Each wave owns exactly one
// feats row, so it zero-fills its own padding tail (no 60MB pre-zero pass).
// ---------------------------------------------------------------------------
__global__ __launch_bounds__(32) void sp_gather(const float* __restrict__ img,
                                                const int*   __restrict__ seg,
                                                float*       __restrict__ feats) {
    const int s    = blockIdx.x;          // 0..195
    const int b    = blockIdx.y;          // 0..63
    const int lane = threadIdx.x;         // wave32

    const int*   segb = seg + (long)b * NPIX;
    const float* imgb = img + (long)b * NCH * NPIX;
    float*       out  = feats + ((long)b * MAX_SEG + s) * FANIN;

    int count = 0;
    for (int p0 = 0; p0 < NPIX; p0 += 32) {
        const int  p     = p0 + lane;
        const bool match = (segb[p] == s);
        const unsigned m32 = (unsigned)__ballot(match);          // wave32: low 32 bits
        const int prefix = __popc(m32 & ((1u << lane) - 1u));
        const int rank   = count + prefix;
        if (match && rank < MAX_PIX) {
            out[0 * MAX_PIX + rank] = imgb[0 * NPIX + p];
            out[1 * MAX_PIX + rank] = imgb[1 * NPIX + p];
            out[2 * MAX_PIX + rank] = imgb[2 * NPIX + p];
        }
        count += __popc(m32);
        if (count >= MAX_PIX) break;     // all later matches truncated anyway
    }

    // Zero the padding tail of this row (count is wave-uniform).
    const int cnt = count < MAX_PIX ? count : MAX_PIX;
#pragma unroll
    for (int c = 0; c < NCH; ++c)
        for (int r = cnt + lane; r < MAX_PIX; r += 32)
            out[c * MAX_PIX + r] = 0.f;
}

// ---------------------------------------------------------------------------
// Kernel 2: fp32 WMMA GEMM.  out[M,N] = feats[M,K] * W[N,K]^T + bias[N]
// M=12544, N=768, K=1200.  Block = 8 waves (4x2), wave = 2x4 16x16 tiles (32x64).
// V_WMMA_F32_16X16X4_F32: A frag[v] = A[m][2h+v], B frag[v] = B[2h+v][n],
// C vgpr v -> row v+8h (h = lane>>4, m/n = lane&15).
// A tile staged via global_load_b128 + ds_store_b128; B tile staged via the
// CDNA5 async global->LDS path (ASYNCcnt), both drained before the barrier.
// ---------------------------------------------------------------------------
__global__ __launch_bounds__(256) void sp_gemm(const float* __restrict__ A,     // [M, K]
                                               const float* __restrict__ Wm,    // [N, K]
                                               const float* __restrict__ bias,  // [N]
                                               float*       __restrict__ out) { // [M, N]
    __shared__ float As[BM][LDST];
    __shared__ float Bs[BN][LDST];

    const int tid  = threadIdx.x;
    const int lane = tid & 31;
    const int wid  = tid >> 5;
    const int wm   = wid >> 1;                 // 0..3 : wave row (32 M each)
    const int wn   = wid & 1;                  // 0..1 : wave col (64 N each)
    const int Mb   = blockIdx.y * BM;
    const int Nb   = blockIdx.x * BN;

    const int m  = lane & 15;
    const int h  = lane >> 4;
    const int kb = h << 1;                     // 0 or 2

    // Staging indices: 128 rows x 12 float4 per tile; 256 threads -> 6 each.
    const int ldr = tid / 2;                   // 0..127 : row handled by this thread
    const int ldq = (tid & 1) * 6;             // 0 or 6 : starting float4 within row
    const unsigned bs_base = (unsigned)(size_t)(&Bs[0][0]);
    const unsigned bs_off0 = bs_base + (unsigned)(ldr * LDST + ldq * 4) * 4u;

    v8f acc[2][4];
#pragma unroll
    for (int i = 0; i < 2; ++i)
#pragma unroll
        for (int j = 0; j < 4; ++j)
            acc[i][j] = (v8f){0.f, 0.f, 0.f, 0.f, 0.f, 0.f, 0.f, 0.f};

    for (int k0 = 0; k0 < FANIN; k0 += KB) {
        // B tile via async global->LDS (6 x b128 per thread, ASYNCcnt-tracked)
        {
            const float* g = Wm + (long)(Nb + ldr) * FANIN + k0 + ldq * 4;
#pragma unroll
            for (int q = 0; q < 6; ++q)
                async_ld_b128(bs_off0 + q * 16u, g + q * 4);
        }
        // A tile via regular load/ds_store (overlaps with async engine)
        {
            const float* g = A + (long)(Mb + ldr) * FANIN + k0 + ldq * 4;
#pragma unroll
            for (int q = 0; q < 6; ++q) {
                const float4 v = *(const float4*)(g + q * 4);
                *(float4*)(&As[ldr][ldq * 4 + q * 4]) = v;
            }
        }
        wait_async0();
        __syncthreads();

#pragma unroll
        for (int ks = 0; ks < KB; ks += 4) {
            v2f afr[2], bfr[4];
#pragma unroll
            for (int tm = 0; tm < 2; ++tm) {
                const int row = wm * 32 + tm * 16 + m;
                afr[tm][0] = As[row][ks + kb];
                afr[tm][1] = As[row][ks + kb + 1];
            }
#pragma unroll
            for (int tn = 0; tn < 4; ++tn) {
                const int col = wn * 64 + tn * 16 + m;
                bfr[tn][0] = Bs[col][ks + kb];
                bfr[tn][1] = Bs[col][ks + kb + 1];
            }
#pragma unroll
            for (int tm = 0; tm < 2; ++tm)
#pragma unroll
                for (int tn = 0; tn < 4; ++tn)
                    acc[tm][tn] = __builtin_amdgcn_wmma_f32_16x16x4_f32(
                        /*neg_a=*/false, afr[tm],
                        /*neg_b=*/false, bfr[tn],
                        /*c_mod=*/(short)0, acc[tm][tn],
                        /*reuse_a=*/false, /*reuse_b=*/false);
        }
        __syncthreads();
    }

    // Epilogue: C layout -> global rows, add bias.
#pragma unroll
    for (int tn = 0; tn < 4; ++tn) {
        const int col = Nb + wn * 64 + tn * 16 + m;
        const float bv = bias[col];
#pragma unroll
        for (int tm = 0; tm < 2; ++tm) {
            const int rbase = Mb + wm * 32 + tm * 16 + h * 8;
#pragma unroll
            for (int v = 0; v < 8; ++v)
                out[(long)(rbase + v) * EMBED + col] = acc[tm][tn][v] + bv;
        }
    }
}

// ---------------------------------------------------------------------------
extern "C" void kernel_launch(void* const* d_in, const int* in_sizes, int n_in,
                              void* d_out, int out_size, void* d_ws, size_t ws_size,
                              hipStream_t stream) {
    const float* img  = (const float*)d_in[0];   // [64, 3, 224, 224]
    const int*   seg  = (const int*)  d_in[1];   // [64, 224, 224]
    const float* Wm   = (const float*)d_in[2];   // [768, 1200]
    const float* bias = (const float*)d_in[3];   // [768]
    float*       out  = (float*)d_out;           // [64, 196, 768]
    float*       feats = (float*)d_ws;           // [12544, 1200] scratch (60.2 MB)

    // 1) gather (writes every element of feats: data + zero tail)
    sp_gather<<<dim3(MAX_SEG, IMG_B), 32, 0, stream>>>(img, seg, feats);

    // 2) fp32 WMMA GEMM: grid covers M=12544 (98 x 128-row tiles), N=768 (6 x 128)
    sp_gemm<<<dim3(EMBED / BN, MROWS / BM), 256, 0, stream>>>(feats, Wm, bias, out);
}